// Model_59906203844677
// MI455X (gfx1250) — compile-verified
//
#include <hip/hip_runtime.h>
#include <math.h>

typedef float v2f __attribute__((ext_vector_type(2)));
typedef float v8f __attribute__((ext_vector_type(8)));

#define B_SZ 4
#define SEQ  2048
#define DM   256
#define NH   4
#define HD   64

// -------- f32 WMMA: D = A(16x4) * B(4x16) + C(16x16), wave32 ----------------
static __device__ __forceinline__ v8f wmma_f32(v2f a, v2f b, v8f c) {
  // 8 args: (neg_a, A, neg_b, B, c_mod, C, reuse_a, reuse_b)
  return __builtin_amdgcn_wmma_f32_16x16x4_f32(false, a, false, b, (short)0, c,
                                               false, false);
}

// reductions across the 16-lane half-groups (match WMMA C-layout row split)
static __device__ __forceinline__ float hmax16(float v) {
  v = fmaxf(v, __shfl_xor(v, 1, 32));
  v = fmaxf(v, __shfl_xor(v, 2, 32));
  v = fmaxf(v, __shfl_xor(v, 4, 32));
  v = fmaxf(v, __shfl_xor(v, 8, 32));
  return v;
}
static __device__ __forceinline__ float hsum16(float v) {
  v += __shfl_xor(v, 1, 32);
  v += __shfl_xor(v, 2, 32);
  v += __shfl_xor(v, 4, 32);
  v += __shfl_xor(v, 8, 32);
  return v;
}

// ---- build transposed weights, PAIR-INTERLEAVED along K --------------------
// WTp[((k>>1)*256 + n)*2 + (k&1)] = W[n][k]   -> GEMM B-fragment = one b64
__global__ void build_mats_kernel(
    const float* __restrict__ qr, const float* __restrict__ qi,
    const float* __restrict__ qj, const float* __restrict__ qk,
    const float* __restrict__ kr, const float* __restrict__ ki,
    const float* __restrict__ kj, const float* __restrict__ kk,
    const float* __restrict__ Wv, const float* __restrict__ Wo,
    float* __restrict__ WqT, float* __restrict__ WkT,
    float* __restrict__ WvT, float* __restrict__ WoT) {
  int tid = blockIdx.x * blockDim.x + threadIdx.x;
  int which = tid >> 16;   // 0:WqT 1:WkT 2:WvT 3:WoT
  int e = tid & 0xFFFF;
  int di = e >> 8;         // input dim  (K of GEMM)
  int dn = e & 255;        // output dim (N of GEMM)
  float v;
  if (which < 2) {
    const int   cmp[4][4] = {{0,1,2,3},{1,0,3,2},{2,3,0,1},{3,2,1,0}};
    const float sgn[4][4] = {{1.f,-1.f,-1.f,-1.f},{1.f,1.f,-1.f,1.f},
                             {1.f,1.f,1.f,-1.f},{1.f,-1.f,1.f,1.f}};
    int R = dn >> 6, p = dn & 63;
    int C = di >> 6, q = di & 63;
    const float* srcs[4];
    if (which == 0) { srcs[0]=qr; srcs[1]=qi; srcs[2]=qj; srcs[3]=qk; }
    else            { srcs[0]=kr; srcs[1]=ki; srcs[2]=kj; srcs[3]=kk; }
    v = sgn[R][C] * srcs[cmp[R][C]][p * 64 + q];
  } else {
    v = ((which == 2) ? Wv : Wo)[dn * 256 + di];
  }
  float* dst = (which == 0) ? WqT : (which == 1) ? WkT
             : (which == 2) ? WvT : WoT;
  dst[(size_t)((di >> 1) * 256 + dn) * 2 + (di & 1)] = v;
}

// ---- per-key threshold: softplus(t) + 0.1 * MLP(density_feat) --------------
__global__ void thr_kernel(const float* __restrict__ feat,
                           const float* __restrict__ thr0,
                           const float* __restrict__ w1,
                           const float* __restrict__ b1,
                           const float* __restrict__ w2,
                           const float* __restrict__ b2,
                           float* __restrict__ thr) {
  int tid = blockIdx.x * blockDim.x + threadIdx.x;   // [0, B*N)
  float f0 = feat[tid * 3 + 0], f1 = feat[tid * 3 + 1], f2 = feat[tid * 3 + 2];
  float d = b2[0];
  #pragma unroll
  for (int j = 0; j < 16; ++j) {
    float h = w1[j * 3 + 0] * f0 + w1[j * 3 + 1] * f1 + w1[j * 3 + 2] * f2 + b1[j];
    h = 0.5f * h * (1.0f + erff(h * 0.70710678118654752f));   // exact GELU
    d += w2[j] * h;
  }
  float t0 = thr0[0];
  float sp = (t0 > 20.f) ? t0 : log1pf(__expf(t0));
  thr[tid] = sp + 0.1f * d;
}

// ---- GEMM: out[M x 256] = X[M x 256] * WTp + bias, wave tile = 64x16 -------
// mode 0: standard [row][col]
// mode 1: Kt pair layout  [bh][k/2][n][2]   (K-projection)
// mode 2: V  pair layout  [bh][n/2][d][2]   (V-projection)
__global__ void gemm_wmma_kernel(const float* __restrict__ X,
                                 const float* __restrict__ WTp,
                                 const float* __restrict__ bias,
                                 float* __restrict__ out, int mode) {
  int wid  = (blockIdx.x * blockDim.x + threadIdx.x) >> 5;
  int lane = threadIdx.x & 31;
  int lh = lane >> 4, ln = lane & 15;
  int rowgrp  = wid >> 4;          // 0..127 (8192 rows / 64)
  int ct      = wid & 15;          // 0..15  (256 cols / 16)
  int rowbase = rowgrp * 64;
  int colbase = ct * 16;

  v8f acc[4] = {};
  for (int k0 = 0; k0 < 256; k0 += 4) {
    int kp = (k0 >> 1) + lh;       // pair index of ka = k0 + 2*lh
    v2f bf = *(const v2f*)(WTp + ((size_t)kp * 256 + colbase + ln) * 2);
    #pragma unroll
    for (int sub = 0; sub < 4; ++sub) {
      int row = rowbase + sub * 16 + ln;
      v2f a = *(const v2f*)(X + (size_t)row * 256 + k0 + 2 * lh);
      acc[sub] = wmma_f32(a, bf, acc[sub]);
    }
  }
  float bv  = bias[colbase + ln];
  int   col = colbase + ln;
  if (mode == 0) {
    #pragma unroll
    for (int sub = 0; sub < 4; ++sub)
      #pragma unroll
      for (int r = 0; r < 8; ++r) {
        int row_m = rowbase + sub * 16 + r + 8 * lh;
        out[(size_t)row_m * 256 + col] = acc[sub][r] + bv;
      }
  } else if (mode == 1) {          // Kt: [(bh*32 + k/2)*4096 + n*2 + (k&1)]
    int h = col >> 6, kki = col & 63;
    #pragma unroll
    for (int sub = 0; sub < 4; ++sub)
      #pragma unroll
      for (int r = 0; r < 8; ++r) {
        int row_m = rowbase + sub * 16 + r + 8 * lh;
        int b = row_m >> 11, n = row_m & 2047;
        out[(size_t)((b * 4 + h) * 32 + (kki >> 1)) * 4096 + n * 2 + (kki & 1)]
            = acc[sub][r] + bv;
      }
  } else {                         // V: [(bh*1024 + n/2)*128 + d*2 + (n&1)]
    int h = col >> 6, d = col & 63;
    #pragma unroll
    for (int sub = 0; sub < 4; ++sub)
      #pragma unroll
      for (int r = 0; r < 8; ++r) {
        int row_m = rowbase + sub * 16 + r + 8 * lh;
        int b = row_m >> 11, n = row_m & 2047;
        out[(size_t)((b * 4 + h) * 1024 + (n >> 1)) * 128 + d * 2 + (n & 1)]
            = acc[sub][r] + bv;
      }
  }
}

// ---- flash attention with spike gating, wave = 16 query rows ---------------
__global__ void attn_kernel(const float* __restrict__ Q,
                            const float* __restrict__ Kt,   // pair layout
                            const float* __restrict__ V,    // pair layout
                            const float* __restrict__ thr,
                            const int* __restrict__ mask,
                            float* __restrict__ Oh) {
  __shared__ float pstag[4 * 256];          // per-wave 16x16 P staging
  int lane = threadIdx.x & 31;
  int w    = threadIdx.x >> 5;              // 0..3
  int lh = lane >> 4, ln = lane & 15;
  int bh  = blockIdx.x >> 5;                // 0..15 = b*4+h
  int blk = blockIdx.x & 31;                // 0..31
  int b = bh >> 2, h = bh & 3;
  int rowbase = blk * 64 + w * 16;

  const float* Qb  = Q  + (size_t)(b * SEQ) * 256 + h * 64;
  const float* Ktb = Kt + (size_t)bh * 131072;   // 32 pair-rows * 4096
  const float* Vb  = V  + (size_t)bh * 131072;   // 1024 pair-rows * 128

  // preload Q tile as 16 A-fragments (hd=64 -> 16 K-steps of 4)
  v2f qa[16];
  {
    const float* qrow = Qb + (size_t)(rowbase + ln) * 256;
    #pragma unroll
    for (int s = 0; s < 16; ++s)
      qa[s] = *(const v2f*)(qrow + 4 * s + 2 * lh);
  }
  float qmf[8];
  #pragma unroll
  for (int r = 0; r < 8; ++r)
    qmf[r] = (float)mask[b * SEQ + rowbase + r + 8 * lh];

  v8f acc[4] = {};
  float rmax[8], rsum[8];
  #pragma unroll
  for (int r = 0; r < 8; ++r) { rmax[r] = -1e30f; rsum[r] = 0.f; }

  float* ps = pstag + w * 256;

  for (int n0 = 0; n0 < SEQ; n0 += 16) {
    // S = Q * K^T  (16x16 tile) -- B-fragments are single b64 loads
    v8f S = {};
    #pragma unroll
    for (int s = 0; s < 16; ++s) {
      int kp = 2 * s + lh;                   // pair index of k = 4s + 2lh
      v2f bf = *(const v2f*)(Ktb + (size_t)kp * 4096 + (n0 + ln) * 2);
      S = wmma_f32(qa[s], bf, S);
    }
    float kmf = (float)mask[b * SEQ + n0 + ln];
    float thv = thr[b * SEQ + n0 + ln];
    if (n0 + 16 < SEQ) {                     // global_prefetch_b8 next tiles
      __builtin_prefetch(Ktb + (n0 + 16) * 2, 0, 0);
      __builtin_prefetch(Vb + (size_t)((n0 + 16) >> 1) * 128, 0, 0);
    }
    // spike gate + mask + online softmax, per C-layout row (vgpr r)
    #pragma unroll
    for (int r = 0; r < 8; ++r) {
      float sc = S[r] * 0.25f;               // 1/sqrt(QQ) = 1/4
      sc = fminf(fmaxf(sc, -6.f), 6.f);
      float g   = 1.0f / (1.0f + __expf(-5.0f * (sc - thv)));
      float mod = sc * (1.0f + 2.0f * g);
      if (qmf[r] * kmf == 0.f) mod = -10000.f;
      mod = fminf(fmaxf(mod, -30.f), 30.f);
      float tm = hmax16(mod);
      float nm = fmaxf(rmax[r], tm);
      float alpha = __expf(rmax[r] - nm);
      float p  = __expf(mod - nm);
      float ts = hsum16(p);
      rsum[r] = rsum[r] * alpha + ts;
      rmax[r] = nm;
      #pragma unroll
      for (int dsub = 0; dsub < 4; ++dsub) acc[dsub][r] *= alpha;
      ps[(r + 8 * lh) * 16 + ln] = p;        // C-layout -> LDS
    }
    asm volatile("s_wait_dscnt 0x0" ::: "memory");   // staging visible (in-order DS)
    // acc += P * V   (P re-read from LDS in A-layout; V fragments are b64)
    #pragma unroll
    for (int s2 = 0; s2 < 4; ++s2) {
      v2f a2 = *(const v2f*)(ps + ln * 16 + 4 * s2 + 2 * lh);
      int npair = (n0 + 4 * s2 + 2 * lh) >> 1;       // nloc even
      const float* vrow = Vb + (size_t)npair * 128;
      #pragma unroll
      for (int dsub = 0; dsub < 4; ++dsub) {
        v2f bf = *(const v2f*)(vrow + (dsub * 16 + ln) * 2);
        acc[dsub] = wmma_f32(a2, bf, acc[dsub]);
      }
    }
    asm volatile("s_wait_dscnt 0x0" ::: "memory");   // reads done before rewrite
  }

  #pragma unroll
  for (int r = 0; r < 8; ++r) {
    float inv = 1.0f / rsum[r];
    int row_m = rowbase + r + 8 * lh;
    float* orow = Oh + (size_t)(b * SEQ + row_m) * 256 + h * 64;
    #pragma unroll
    for (int dsub = 0; dsub < 4; ++dsub)
      orow[dsub * 16 + ln] = acc[dsub][r] * inv;
  }
}

extern "C" void kernel_launch(void* const* d_in, const int* in_sizes, int n_in,
                              void* d_out, int out_size, void* d_ws,
                              size_t ws_size, hipStream_t stream) {
  (void)in_sizes; (void)n_in; (void)out_size; (void)ws_size;
  const float* query = (const float*)d_in[0];
  const float* key   = (const float*)d_in[1];
  const float* value = (const float*)d_in[2];
  const float* dfeat = (const float*)d_in[3];
  const int*   maskp = (const int*)d_in[4];
  const float* Wq_r = (const float*)d_in[5];
  const float* Wq_i = (const float*)d_in[6];
  const float* Wq_j = (const float*)d_in[7];
  const float* Wq_k = (const float*)d_in[8];
  const float* bq   = (const float*)d_in[9];
  const float* Wk_r = (const float*)d_in[10];
  const float* Wk_i = (const float*)d_in[11];
  const float* Wk_j = (const float*)d_in[12];
  const float* Wk_k = (const float*)d_in[13];
  const float* bk   = (const float*)d_in[14];
  const float* Wv   = (const float*)d_in[15];
  const float* bv   = (const float*)d_in[16];
  const float* Wo   = (const float*)d_in[17];
  const float* bo   = (const float*)d_in[18];
  const float* th0  = (const float*)d_in[19];
  const float* dm_w1 = (const float*)d_in[20];
  const float* dm_b1 = (const float*)d_in[21];
  const float* dm_w2 = (const float*)d_in[22];
  const float* dm_b2 = (const float*)d_in[23];

  float* ws   = (float*)d_ws;
  float* WqT  = ws;                 // 65536
  float* WkT  = ws + 65536;         // 65536
  float* WvT  = ws + 131072;        // 65536
  float* WoT  = ws + 196608;        // 65536
  float* Qp   = ws + 262144;        // 2097152
  float* Ktp  = ws + 2359296;       // 2097152  pair layout [bh][k/2][n][2]
  float* Vp   = ws + 4456448;       // 2097152  pair layout [bh][n/2][d][2]
  float* Ohp  = ws + 6553600;       // 2097152
  float* thrp = ws + 8650752;       // 8192   (total 34.6 MB)

  build_mats_kernel<<<dim3(1024), dim3(256), 0, stream>>>(
      Wq_r, Wq_i, Wq_j, Wq_k, Wk_r, Wk_i, Wk_j, Wk_k, Wv, Wo,
      WqT, WkT, WvT, WoT);
  thr_kernel<<<dim3(32), dim3(256), 0, stream>>>(
      dfeat, th0, dm_w1, dm_b1, dm_w2, dm_b2, thrp);
  gemm_wmma_kernel<<<dim3(256), dim3(256), 0, stream>>>(query, WqT, bq, Qp, 0);
  gemm_wmma_kernel<<<dim3(256), dim3(256), 0, stream>>>(key,   WkT, bk, Ktp, 1);
  gemm_wmma_kernel<<<dim3(256), dim3(256), 0, stream>>>(value, WvT, bv, Vp, 2);
  attn_kernel<<<dim3(512), dim3(128), 0, stream>>>(Qp, Ktp, Vp, thrp, maskp, Ohp);
  gemm_wmma_kernel<<<dim3(256), dim3(256), 0, stream>>>(Ohp, WoT, bo,
                                                        (float*)d_out, 0);
}